// Head_10840497455175
// MI455X (gfx1250) — compile-verified
//
#include <hip/hip_runtime.h>
#include <stdint.h>

typedef float v2f __attribute__((ext_vector_type(2)));
typedef float v8f __attribute__((ext_vector_type(8)));

#define T_DIM 2048
#define C_DIM 1024
#define H_DIM 64
#define B_DIM 8
#define SCALE 0.03125f   // C**-0.5 = 1024**-0.5

static __device__ __forceinline__ v8f wmma_f32(v2f a, v2f b, v8f c) {
    // V_WMMA_F32_16X16X4_F32 : D(16x16 f32) = A(16x4 f32) * B(4x16 f32) + C
    return __builtin_amdgcn_wmma_f32_16x16x4_f32(
        false, a, false, b, (short)0, c, false, false);
}

// Async DMA one 16-byte packet per lane into LDS (ASYNCcnt-tracked).
static __device__ __forceinline__ void async_b128(uint32_t lds_byte_addr,
                                                  const float* gaddr) {
    asm volatile("global_load_async_to_lds_b128 %0, %1, off"
                 :: "v"(lds_byte_addr), "v"(gaddr) : "memory");
}
static __device__ __forceinline__ void wait_async0() {
    asm volatile("s_wait_asynccnt 0" ::: "memory");
}
static __device__ __forceinline__ uint32_t lds_addr(const void* p) {
    // generic pointer to LDS: low 32 bits are the wave-relative LDS address
    return (uint32_t)(uintptr_t)p;
}

// ---------------------------------------------------------------------------
// W transpose: Wt[m][h][c] = W_m[c][h]   (m: 0=q,1=k,2=v)  768 KB total
// ---------------------------------------------------------------------------
__global__ __launch_bounds__(256)
void wtrans_kernel(const float* __restrict__ Wk, const float* __restrict__ Wq,
                   const float* __restrict__ Wv, float* __restrict__ wt)
{
    const int m = blockIdx.y;
    const float* W = (m == 0) ? Wq : (m == 1 ? Wk : Wv);
    const int idx = blockIdx.x * 256 + threadIdx.x;   // 0..65535
    const int c = idx >> 6, h = idx & 63;
    wt[(size_t)m * (C_DIM * H_DIM) + (size_t)h * C_DIM + c] = W[idx];
}

// ---------------------------------------------------------------------------
// Projection: 16 rows of X @ W (K-major staged operands, async double-buffer)
// blockIdx.x: row tile over B*T; blockIdx.y: 0=q, 1=k, 2=v(transposed out)
// ---------------------------------------------------------------------------
#define XSW 68   // 16B-aligned row stride, banks 4*ln -> conflict-free
#define WSW 68

__global__ __launch_bounds__(32)
void proj_kernel(const float* __restrict__ x,
                 const float* __restrict__ wt,   // [3][64][1024]
                 float* __restrict__ outq,
                 float* __restrict__ outk,
                 float* __restrict__ outv)       // outv transposed [B][H][T]
{
    __shared__ float Xs[2][16 * XSW];
    __shared__ float Ws[2][64 * WSW];

    const int lane = threadIdx.x;
    const int ln   = lane & 15;
    const int h2   = (lane < 16) ? 0 : 2;
    const int mofs = (lane < 16) ? 0 : 8;
    const int rowbase = blockIdx.x * 16;
    const int my = blockIdx.y;
    const float* Wm = wt + (size_t)my * (C_DIM * H_DIM);   // [64][1024] h-major

    const uint32_t xs0 = lds_addr(&Xs[0][0]), xs1 = lds_addr(&Xs[1][0]);
    const uint32_t ws0 = lds_addr(&Ws[0][0]), ws1 = lds_addr(&Ws[1][0]);

    v8f acc[4];
#pragma unroll
    for (int nt = 0; nt < 4; nt++)
#pragma unroll
        for (int r = 0; r < 8; r++) acc[nt][r] = 0.0f;

    // ---- issue helper: partial unroll keeps async address pressure tiny ----
#define PROJ_ISSUE(kb, xsb, wsb)                                               \
    do {                                                                       \
        _Pragma("unroll 2")                                                    \
        for (int i = 0; i < 8; i++) {                                          \
            int idx = i * 128 + lane * 4;                                      \
            int r = idx >> 6, c = idx & 63;                                    \
            async_b128((xsb) + (uint32_t)(r * XSW + c) * 4,                    \
                       &x[(size_t)(rowbase + r) * C_DIM + (kb) + c]);          \
        }                                                                      \
        _Pragma("unroll 2")                                                    \
        for (int i = 0; i < 32; i++) {                                         \
            int idx = i * 128 + lane * 4;                                      \
            int n = idx >> 6, kc = idx & 63;                                   \
            async_b128((wsb) + (uint32_t)(n * WSW + kc) * 4,                   \
                       &Wm[(size_t)n * C_DIM + (kb) + kc]);                    \
        }                                                                      \
    } while (0)

    PROJ_ISSUE(0, xs0, ws0);

    for (int cidx = 0; cidx < C_DIM / 64; cidx++) {
        const int buf = cidx & 1;
        const float* XsL = Xs[buf];
        const float* WsL = Ws[buf];

        wait_async0();                         // current chunk landed
        if (cidx + 1 < C_DIM / 64) {           // prefetch next chunk
            if (buf == 0) PROJ_ISSUE((cidx + 1) * 64, xs1, ws1);
            else          PROJ_ISSUE((cidx + 1) * 64, xs0, ws0);
        }

#pragma unroll
        for (int kk = 0; kk < 64; kk += 4) {
            v2f a = *(const v2f*)&XsL[ln * XSW + kk + h2];
#pragma unroll
            for (int nt = 0; nt < 4; nt++) {
                v2f b = *(const v2f*)&WsL[(nt * 16 + ln) * WSW + kk + h2];
                acc[nt] = wmma_f32(a, b, acc[nt]);
            }
        }
    }
#undef PROJ_ISSUE

    if (my == 2) {  // V: write transposed [B][H][T]
        const int bidx = rowbase / T_DIM;
        const int t0   = rowbase % T_DIM;
#pragma unroll
        for (int nt = 0; nt < 4; nt++)
#pragma unroll
            for (int r = 0; r < 8; r++)
                outv[((size_t)bidx * H_DIM + nt * 16 + ln) * T_DIM + t0 + r + mofs] =
                    acc[nt][r];
    } else {
        float* out = (my == 0) ? outq : outk;
#pragma unroll
        for (int nt = 0; nt < 4; nt++)
#pragma unroll
            for (int r = 0; r < 8; r++)
                out[(size_t)(rowbase + r + mofs) * H_DIM + nt * 16 + ln] = acc[nt][r];
    }
}

// ---------------------------------------------------------------------------
// Fused causal flash attention, async double-buffered K/V tiles.
// blockIdx.x = query tile (128), blockIdx.y = batch (8). One wave per block.
// ---------------------------------------------------------------------------
#define KSW 68   // K tile row stride (row-major [key][feat])
#define VTW 20   // V tile row stride (feat-major [feat][key]), 80B rows
#define PSW 20

__global__ __launch_bounds__(32)
void attn_kernel(const float* __restrict__ q,     // [B][T][H]
                 const float* __restrict__ k,     // [B][T][H]
                 const float* __restrict__ vt,    // [B][H][T]  (transposed)
                 float* __restrict__ out)
{
    __shared__ float Ks[2][16 * KSW];
    __shared__ float Vs[2][64 * VTW];
    __shared__ float Ps[16 * PSW];

    const int lane = threadIdx.x;
    const int ln   = lane & 15;
    const int h2   = (lane < 16) ? 0 : 2;
    const int mofs = (lane < 16) ? 0 : 8;
    const int b    = blockIdx.y;
    const int qt   = blockIdx.x;
    const int qbase = qt * 16;

    const float* Qp  = q  + ((size_t)b * T_DIM + qbase) * H_DIM;
    const float* Kp  = k  + (size_t)b * T_DIM * H_DIM;
    const float* Vbp = vt + (size_t)b * H_DIM * T_DIM;

    const uint32_t ks0 = lds_addr(&Ks[0][0]), ks1 = lds_addr(&Ks[1][0]);
    const uint32_t vs0 = lds_addr(&Vs[0][0]), vs1 = lds_addr(&Vs[1][0]);

#define ATTN_ISSUE(jb, ksb, vsb)                                               \
    do {                                                                       \
        _Pragma("unroll 2")                                                    \
        for (int i = 0; i < 8; i++) {                                          \
            int idx = i * 128 + lane * 4;                                      \
            int r = idx >> 6, c = idx & 63;                                    \
            async_b128((ksb) + (uint32_t)(r * KSW + c) * 4,                    \
                       &Kp[(size_t)((jb) + r) * H_DIM + c]);                   \
        }                                                                      \
        _Pragma("unroll 2")                                                    \
        for (int i = 0; i < 8; i++) {                                          \
            int idx = i * 128 + lane * 4;                                      \
            int h = idx >> 4, c = idx & 15;                                    \
            async_b128((vsb) + (uint32_t)(h * VTW + c) * 4,                    \
                       &Vbp[(size_t)h * T_DIM + (jb) + c]);                    \
        }                                                                      \
    } while (0)

    // Preload Q fragments in A layout: lane ln holds row ln, K pair at h2
    v2f qa[16];
#pragma unroll
    for (int st = 0; st < 16; st++)
        qa[st] = *(const v2f*)&Qp[(size_t)ln * H_DIM + st * 4 + h2];

    v8f acc[4];
    float m_st[8], l_st[8];
#pragma unroll
    for (int nt = 0; nt < 4; nt++)
#pragma unroll
        for (int r = 0; r < 8; r++) acc[nt][r] = 0.0f;
#pragma unroll
    for (int r = 0; r < 8; r++) { m_st[r] = -__builtin_inff(); l_st[r] = 0.0f; }

    ATTN_ISSUE(0, ks0, vs0);

    for (int jt = 0; jt <= qt; jt++) {
        const int buf = jt & 1;
        const float* KsL = Ks[buf];
        const float* VsL = Vs[buf];

        wait_async0();                        // tile jt landed in LDS
        if (jt < qt) {                        // prefetch tile jt+1
            if (buf == 0) ATTN_ISSUE((jt + 1) * 16, ks1, vs1);
            else          ATTN_ISSUE((jt + 1) * 16, ks0, vs0);
        }

        // S = Q @ K^T
        v8f s;
#pragma unroll
        for (int r = 0; r < 8; r++) s[r] = 0.0f;
#pragma unroll
        for (int st = 0; st < 16; st++) {
            v2f bf = *(const v2f*)&KsL[ln * KSW + st * 4 + h2];
            s = wmma_f32(qa[st], bf, s);
        }

        // scale + causal mask (only diagonal tile masks)
#pragma unroll
        for (int r = 0; r < 8; r++) {
            float val = s[r] * SCALE;
            if (jt == qt && ln > r + mofs) val = -__builtin_inff();
            s[r] = val;
        }

        // online softmax: row r+mofs lives in one 16-lane half
#pragma unroll
        for (int r = 0; r < 8; r++) {
            float mt = s[r];
            mt = fmaxf(mt, __shfl_xor(mt, 1));
            mt = fmaxf(mt, __shfl_xor(mt, 2));
            mt = fmaxf(mt, __shfl_xor(mt, 4));
            mt = fmaxf(mt, __shfl_xor(mt, 8));
            float m_new = fmaxf(m_st[r], mt);
            float alpha = __expf(m_st[r] - m_new);
            float pv    = __expf(s[r] - m_new);
            float rs = pv;
            rs += __shfl_xor(rs, 1);
            rs += __shfl_xor(rs, 2);
            rs += __shfl_xor(rs, 4);
            rs += __shfl_xor(rs, 8);
            l_st[r] = l_st[r] * alpha + rs;
            m_st[r] = m_new;
#pragma unroll
            for (int nt = 0; nt < 4; nt++) acc[nt][r] *= alpha;
            Ps[(r + mofs) * PSW + ln] = pv;   // C-layout -> A-layout bounce
        }
        __syncthreads();

        // O += P @ V   (B fragment pairs contiguous in feat-major V tile)
#pragma unroll
        for (int st = 0; st < 4; st++) {
            v2f a = *(const v2f*)&Ps[ln * PSW + st * 4 + h2];
#pragma unroll
            for (int nt = 0; nt < 4; nt++) {
                v2f bf = *(const v2f*)&VsL[(nt * 16 + ln) * VTW + st * 4 + h2];
                acc[nt] = wmma_f32(a, bf, acc[nt]);
            }
        }
        __syncthreads();
    }
#undef ATTN_ISSUE

    float* Op = out + ((size_t)b * T_DIM + qbase) * H_DIM;
#pragma unroll
    for (int nt = 0; nt < 4; nt++)
#pragma unroll
        for (int r = 0; r < 8; r++)
            Op[(size_t)(r + mofs) * H_DIM + nt * 16 + ln] = acc[nt][r] / l_st[r];
}

extern "C" void kernel_launch(void* const* d_in, const int* in_sizes, int n_in,
                              void* d_out, int out_size, void* d_ws, size_t ws_size,
                              hipStream_t stream) {
    // setup_inputs order: x, Wk, Wq, Wv
    const float* x  = (const float*)d_in[0];
    const float* Wk = (const float*)d_in[1];
    const float* Wq = (const float*)d_in[2];
    const float* Wv = (const float*)d_in[3];

    float* wsf = (float*)d_ws;
    const size_t BTH = (size_t)B_DIM * T_DIM * H_DIM;   // 1,048,576 floats
    float* qbuf = wsf;                  // [B][T][H]
    float* kbuf = wsf + BTH;            // [B][T][H]
    float* vbuf = wsf + 2 * BTH;        // [B][H][T] transposed
    float* wt   = wsf + 3 * BTH;        // [3][64][1024]

    dim3 tgrid((C_DIM * H_DIM) / 256, 3);
    wtrans_kernel<<<tgrid, 256, 0, stream>>>(Wk, Wq, Wv, wt);

    dim3 pgrid((B_DIM * T_DIM) / 16, 3);
    proj_kernel<<<pgrid, 32, 0, stream>>>(x, wt, qbuf, kbuf, vbuf);

    dim3 agrid(T_DIM / 16, B_DIM);
    attn_kernel<<<agrid, 32, 0, stream>>>(qbuf, kbuf, vbuf, (float*)d_out);
}